// QuixerCore_65103114273460
// MI455X (gfx1250) — compile-verified
//
#include <hip/hip_runtime.h>
#include <math.h>

// CDNA5 / gfx1250 fused Quixer kernel.
// Single workgroup (256 threads = 8 waves of 32), state vectors in LDS.
// WMMA f32 16x16x4 for the angles GEMM; RY gate sweeps + reductions for the rest.

#define DIMQ   2048     // 2^11 amplitudes
#define NQ     11
#define NPQC   44
#define DMODEL 512

typedef __attribute__((ext_vector_type(2))) float v2f;
typedef __attribute__((ext_vector_type(8))) float v8f;

__device__ __forceinline__ float block_sum(float v, float* wred, int tid) {
    // wave32 butterfly reduce, then cross-wave via LDS
    for (int o = 16; o; o >>= 1) v += __shfl_xor(v, o, 32);
    if ((tid & 31) == 0) wred[tid >> 5] = v;
    __syncthreads();
    if (tid == 0) {
        float r = 0.f;
        for (int w = 0; w < 8; ++w) r += wred[w];
        wred[0] = r;
    }
    __syncthreads();
    float r = wred[0];
    __syncthreads();   // wred reused by later calls
    return r;
}

// RY(theta) on qubit with bit position b (bit 10-q of the amplitude index):
// new0 = c*v0 - s*v1 ; new1 = s*v0 + c*v1  (real gate: re/im independent)
__device__ __forceinline__ void apply_ry(float2* st, int b, float c, float s, int tid) {
    #pragma unroll
    for (int j = 0; j < 4; ++j) {
        int p  = tid + 256 * j;                       // pair index 0..1023
        int i0 = ((p >> b) << (b + 1)) | (p & ((1 << b) - 1));
        int i1 = i0 | (1 << b);
        float2 v0 = st[i0], v1 = st[i1];
        st[i0] = make_float2(c * v0.x - s * v1.x, c * v0.y - s * v1.y);
        st[i1] = make_float2(s * v0.x + c * v1.x, s * v0.y + c * v1.y);
    }
}

__global__ __launch_bounds__(256) void quixer_fused_kernel(
    const float* __restrict__ tokens,   // [32][512]
    const float* __restrict__ Wang,     // [44][512]
    const float* __restrict__ bang,     // [44]
    const float* __restrict__ qsvt,     // [3]
    const float* __restrict__ lre,      // [32]
    const float* __restrict__ lim,      // [32]
    const float* __restrict__ ffp,      // [44]
    float* __restrict__ out)            // [33]
{
    __shared__ float2 s_mono[DIMQ];     // 16 KB
    __shared__ float2 s_work[DIMQ];     // 16 KB
    __shared__ float2 s_lcs [DIMQ];     // 16 KB
    __shared__ float  s_ang[4 * NPQC];  // angles for tokens 0..3
    __shared__ float  s_wred[8];
    __shared__ float  s_scal[2];
    __shared__ float  s_lcu[8];         // (re,im) per token 0..3

    const int tid  = threadIdx.x;
    const int lane = tid & 31;
    const int wv   = tid >> 5;

    // ---------- Phase A: angles[0:4][44] = tokens[0:4] @ W^T + b  (WMMA) ----------
    // Waves 0..2 each own one 16-wide N tile; K=512 in 128 chained 16x16x4 WMMAs.
    // Wave-uniform branch => EXEC stays all-ones inside each WMMA wave.
    if (wv < 3) {
        const int n0   = wv * 16;
        const int half = lane >> 4;          // 0: K pair (0,1), 1: K pair (2,3)
        const int off  = half << 1;
        const int m    = lane & 15;          // A row (tokens rows 0..15 all valid)
        const int n    = n0 + (lane & 15);   // B col
        const int nc   = (n < NPQC) ? n : (NPQC - 1);  // clamp OOB (results unused)
        const float* arow = tokens + m  * DMODEL;
        const float* brow = Wang   + nc * DMODEL;
        v8f acc = {};
        for (int k0 = 0; k0 < DMODEL; k0 += 4) {
            v2f av, bv;
            av.x = arow[k0 + off]; av.y = arow[k0 + off + 1];
            bv.x = brow[k0 + off]; bv.y = brow[k0 + off + 1];
            acc = __builtin_amdgcn_wmma_f32_16x16x4_f32(
                      false, av, false, bv, (short)0, acc, false, false);
        }
        // D layout: lanes 0-15 VGPR v -> row M=v, col N=lane. Rows 0..3 wanted.
        if (half == 0 && n < NPQC) {
            float bb = bang[n];
            s_ang[0 * NPQC + n] = acc[0] + bb;
            s_ang[1 * NPQC + n] = acc[1] + bb;
            s_ang[2 * NPQC + n] = acc[2] + bb;
            s_ang[3 * NPQC + n] = acc[3] + bb;
        }
    } else if (wv == 3) {
        // LCU normalization constant: sum |lcu| over all 32 tokens
        float re = lre[lane], im = lim[lane];
        float v  = sqrtf(re * re + im * im);
        for (int o = 16; o; o >>= 1) v += __shfl_xor(v, o, 32);
        if (lane == 0) s_scal[0] = fmaxf(v, 1e-8f);
    }
    __syncthreads();

    if (tid < 4) {
        float s = s_scal[0];
        s_lcu[tid * 2 + 0] = lre[tid] / s;
        s_lcu[tid * 2 + 1] = lim[tid] / s;
    }

    // mono = e0 ; acc registers = qsvt[0]*e0 (8 strided elements per thread)
    const float q0 = qsvt[0], q1 = qsvt[1], q2 = qsvt[2];
    float accR[8], accI[8];
    #pragma unroll
    for (int j = 0; j < 8; ++j) {
        int e = tid + 256 * j;
        s_mono[e] = make_float2((e == 0) ? 1.f : 0.f, 0.f);
        accR[j] = (e == 0) ? q0 : 0.f;
        accI[j] = 0.f;
    }
    __syncthreads();

    // ---------- Phase B: QSVT degree-2 iteration ----------
    for (int deg = 1; deg <= 2; ++deg) {
        // norm of mono
        float part = 0.f;
        #pragma unroll
        for (int j = 0; j < 8; ++j) {
            float2 v = s_mono[tid + 256 * j];
            part += v.x * v.x + v.y * v.y;
        }
        float nrm   = sqrtf(block_sum(part, s_wred, tid));
        int   small = (nrm < 1e-8f);           // block-uniform
        float inv   = small ? 0.f : (1.f / nrm);
        #pragma unroll
        for (int j = 0; j < 8; ++j) {
            int e = tid + 256 * j;
            float2 v = s_mono[e];
            s_mono[e] = small ? make_float2((e == 0) ? 1.f : 0.f, 0.f)
                              : make_float2(v.x * inv, v.y * inv);
            s_lcs[e]  = make_float2(0.f, 0.f);
        }
        __syncthreads();

        for (int t = 0; t < 4; ++t) {
            #pragma unroll
            for (int j = 0; j < 8; ++j) {
                int e = tid + 256 * j;
                s_work[e] = s_mono[e];
            }
            __syncthreads();
            // 22 RY gates: params 0..10 (qubits 0..10), then 22..32 (qubits 0..10)
            for (int g = 0; g < 22; ++g) {
                int   i    = (g < 11) ? g : (g - 11);
                int   pidx = (g < 11) ? g : (22 + i);
                int   b    = 10 - i;                   // qubit 0 = MSB of index
                float th   = 0.5f * s_ang[t * NPQC + pidx];
                float cc = cosf(th), ss = sinf(th);
                apply_ry(s_work, b, cc, ss, tid);
                __syncthreads();
            }
            // lcsum += lcu_t * work   (complex scalar * vector)
            float lr = s_lcu[t * 2], li = s_lcu[t * 2 + 1];
            #pragma unroll
            for (int j = 0; j < 8; ++j) {
                int e = tid + 256 * j;
                float2 w = s_work[e];
                float2 a = s_lcs[e];
                a.x += lr * w.x - li * w.y;
                a.y += lr * w.y + li * w.x;
                s_lcs[e] = a;
            }
            __syncthreads();
        }
        // mono = (small?1:nrm) * lcsum ; acc += qsvt[deg] * mono
        float fac = small ? 1.f : nrm;
        float qk  = (deg == 1) ? q1 : q2;
        #pragma unroll
        for (int j = 0; j < 8; ++j) {
            int e = tid + 256 * j;
            float2 v = s_lcs[e];
            v.x *= fac; v.y *= fac;
            s_mono[e] = v;
            accR[j] += qk * v.x;
            accI[j] += qk * v.y;
        }
        __syncthreads();
    }

    // ---------- Phase C: normalize acc, apply feed-forward unitary ----------
    float pn  = fmaxf(fabsf(q0) + fabsf(q1) + fabsf(q2), 1e-8f);
    float ipn = 1.f / pn;
    float part = 0.f;
    #pragma unroll
    for (int j = 0; j < 8; ++j) {
        accR[j] *= ipn; accI[j] *= ipn;
        part += accR[j] * accR[j] + accI[j] * accI[j];
    }
    float sn    = sqrtf(block_sum(part, s_wred, tid));
    int   small = (sn < 1e-8f);
    float inv   = small ? 0.f : (1.f / sn);
    #pragma unroll
    for (int j = 0; j < 8; ++j) {
        int e = tid + 256 * j;
        s_work[e] = small ? make_float2((e == 0) ? 1.f : 0.f, 0.f)
                          : make_float2(accR[j] * inv, accI[j] * inv);
    }
    __syncthreads();
    for (int g = 0; g < 22; ++g) {
        int   i    = (g < 11) ? g : (g - 11);
        int   pidx = (g < 11) ? g : (22 + i);
        int   b    = 10 - i;
        float th   = 0.5f * ffp[pidx];
        float cc = cosf(th), ss = sinf(th);
        apply_ry(s_work, b, cc, ss, tid);
        __syncthreads();
    }

    // ---------- Phase D: Pauli expectations <X>,<Y>,<Z> per qubit ----------
    for (int q = 0; q < NQ; ++q) {
        int b = 10 - q;
        float px = 0.f, py = 0.f, pz = 0.f;
        #pragma unroll
        for (int j = 0; j < 4; ++j) {
            int p  = tid + 256 * j;
            int i0 = ((p >> b) << (b + 1)) | (p & ((1 << b) - 1));
            int i1 = i0 | (1 << b);
            float2 v0 = s_work[i0], v1 = s_work[i1];
            px += 2.f * (v0.x * v1.x + v0.y * v1.y);            // 2 Re(conj(s0) s1)
            py += 2.f * (v0.x * v1.y - v0.y * v1.x);            // 2 Im(conj(s0) s1)
            pz += (v0.x * v0.x + v0.y * v0.y) - (v1.x * v1.x + v1.y * v1.y);
        }
        float sx = block_sum(px, s_wred, tid);
        float sy = block_sum(py, s_wred, tid);
        float sz = block_sum(pz, s_wred, tid);
        if (tid == 0) {
            out[q * 3 + 0] = sx;
            out[q * 3 + 1] = sy;
            out[q * 3 + 2] = sz;
        }
    }
}

extern "C" void kernel_launch(void* const* d_in, const int* in_sizes, int n_in,
                              void* d_out, int out_size, void* d_ws, size_t ws_size,
                              hipStream_t stream) {
    (void)in_sizes; (void)n_in; (void)out_size; (void)d_ws; (void)ws_size;
    const float* tokens = (const float*)d_in[0];
    const float* Wang   = (const float*)d_in[1];
    const float* bang   = (const float*)d_in[2];
    const float* qsvt   = (const float*)d_in[3];
    const float* lre    = (const float*)d_in[4];
    const float* lim    = (const float*)d_in[5];
    const float* ffp    = (const float*)d_in[6];
    quixer_fused_kernel<<<1, 256, 0, stream>>>(tokens, Wang, bang, qsvt,
                                               lre, lim, ffp, (float*)d_out);
}